// GaussianRasterizer_24833500906049
// MI455X (gfx1250) — compile-verified
//
#include <hip/hip_runtime.h>
#include <math.h>

// ---------------- problem constants (match reference) ----------------
static constexpr int   GN     = 2048;
static constexpr int   IMG_H  = 128;
static constexpr int   IMG_W  = 128;
static constexpr int   HW     = IMG_H * IMG_W;
static constexpr float TANFOV = 0.5f;
static constexpr float FOCAL  = IMG_W / (2.0f * TANFOV); // 128

typedef __attribute__((ext_vector_type(2))) float v2f;
typedef __attribute__((ext_vector_type(8))) float v8f;

// ws layout (floats):
//   [0, GN)          keys (tz or +inf)
//   [GN, 10*GN)      unsorted params U[9][GN]  (A,B,C,op,px,py,r,g,b)
//   [10*GN, 19*GN)   sorted params  S[9][GN]
// ---------------------------------------------------------------------

// ------------------- kernel 1: per-gaussian preprocess ---------------
__global__ void gsplat_preprocess(const float* __restrict__ means3d,
                                  const float* __restrict__ colors,
                                  const float* __restrict__ opac,
                                  const float* __restrict__ scales,
                                  const float* __restrict__ rots,
                                  const float* __restrict__ Vm,   // 4x4 row-major
                                  const float* __restrict__ Pm,   // 4x4 row-major
                                  float* __restrict__ keys,
                                  float* __restrict__ U,
                                  float* __restrict__ out_radii)
{
    int i = blockIdx.x * blockDim.x + threadIdx.x;
    if (i >= GN) return;

    // quaternion -> rotation
    float qr = rots[4*i+0], qx = rots[4*i+1], qy = rots[4*i+2], qz = rots[4*i+3];
    float qn = sqrtf(qr*qr + qx*qx + qy*qy + qz*qz) + 1e-12f;
    qr /= qn; qx /= qn; qy /= qn; qz /= qn;
    float R00 = 1.f - 2.f*(qy*qy + qz*qz), R01 = 2.f*(qx*qy - qr*qz), R02 = 2.f*(qx*qz + qr*qy);
    float R10 = 2.f*(qx*qy + qr*qz), R11 = 1.f - 2.f*(qx*qx + qz*qz), R12 = 2.f*(qy*qz - qr*qx);
    float R20 = 2.f*(qx*qz - qr*qy), R21 = 2.f*(qy*qz + qr*qx), R22 = 1.f - 2.f*(qx*qx + qy*qy);

    float s0 = scales[3*i+0], s1 = scales[3*i+1], s2 = scales[3*i+2];
    float v0 = s0*s0, v1 = s1*s1, v2 = s2*s2;       // SCALE_MOD = 1
    // cov3d = R diag(v) R^T  (symmetric, 6 uniques)
    float c00 = R00*R00*v0 + R01*R01*v1 + R02*R02*v2;
    float c01 = R00*R10*v0 + R01*R11*v1 + R02*R12*v2;
    float c02 = R00*R20*v0 + R01*R21*v1 + R02*R22*v2;
    float c11 = R10*R10*v0 + R11*R11*v1 + R12*R12*v2;
    float c12 = R10*R20*v0 + R11*R21*v1 + R12*R22*v2;
    float c22 = R20*R20*v0 + R21*R21*v1 + R22*R22*v2;

    float m0 = means3d[3*i+0], m1 = means3d[3*i+1], m2 = means3d[3*i+2];
    // view-space position
    float pv0 = Vm[0]*m0 + Vm[1]*m1 + Vm[2]*m2  + Vm[3];
    float pv1 = Vm[4]*m0 + Vm[5]*m1 + Vm[6]*m2  + Vm[7];
    float pv2 = Vm[8]*m0 + Vm[9]*m1 + Vm[10]*m2 + Vm[11];
    // homogeneous projection
    float ph0 = Pm[0]*m0  + Pm[1]*m1  + Pm[2]*m2  + Pm[3];
    float ph1 = Pm[4]*m0  + Pm[5]*m1  + Pm[6]*m2  + Pm[7];
    float ph3 = Pm[12]*m0 + Pm[13]*m1 + Pm[14]*m2 + Pm[15];
    float invw = 1.0f / (ph3 + 1e-7f);
    float pr0 = ph0 * invw, pr1 = ph1 * invw;

    float tz = pv2;
    float lim = 1.3f * TANFOV;
    float tx = fminf(fmaxf(pv0 / tz, -lim), lim) * tz;
    float ty = fminf(fmaxf(pv1 / tz, -lim), lim) * tz;

    float itz = 1.0f / tz;
    float J00 = FOCAL * itz, J02 = -FOCAL * tx * itz * itz;
    float J11 = FOCAL * itz, J12 = -FOCAL * ty * itz * itz;
    // Tm = J @ V[:3,:3]
    float T00 = J00*Vm[0] + J02*Vm[8];
    float T01 = J00*Vm[1] + J02*Vm[9];
    float T02 = J00*Vm[2] + J02*Vm[10];
    float T10 = J11*Vm[4] + J12*Vm[8];
    float T11 = J11*Vm[5] + J12*Vm[9];
    float T12 = J11*Vm[6] + J12*Vm[10];
    // cov2d = Tm cov3d Tm^T
    float w00 = T00*c00 + T01*c01 + T02*c02;
    float w01 = T00*c01 + T01*c11 + T02*c12;
    float w02 = T00*c02 + T01*c12 + T02*c22;
    float x00 = T10*c00 + T11*c01 + T12*c02;
    float x01 = T10*c01 + T11*c11 + T12*c12;
    float x02 = T10*c02 + T11*c12 + T12*c22;
    float a = w00*T00 + w01*T01 + w02*T02 + 0.3f;
    float b = w00*T10 + w01*T11 + w02*T12;
    float c = x00*T10 + x01*T11 + x02*T12 + 0.3f;

    float det = a*c - b*b;
    bool valid = (tz > 0.2f) && (det > 0.0f);
    float inv_det = valid ? (1.0f / det) : 0.0f;
    float conA =  c * inv_det;
    float conB = -b * inv_det;
    float conC =  a * inv_det;

    float mid = 0.5f * (a + c);
    float lam = mid + sqrtf(fmaxf(0.1f, mid*mid - det));
    float radiusf = ceilf(3.0f * sqrtf(lam));

    float px = ((pr0 + 1.0f) * (float)IMG_W - 1.0f) * 0.5f;
    float py = ((pr1 + 1.0f) * (float)IMG_H - 1.0f) * 0.5f;

    keys[i]      = valid ? tz : __builtin_inff();
    U[0*GN + i]  = conA;
    U[1*GN + i]  = conB;
    U[2*GN + i]  = conC;
    U[3*GN + i]  = valid ? opac[i] : 0.0f;   // invalid -> alpha forced 0
    U[4*GN + i]  = px;
    U[5*GN + i]  = py;
    U[6*GN + i]  = colors[3*i+0];
    U[7*GN + i]  = colors[3*i+1];
    U[8*GN + i]  = colors[3*i+2];

    out_radii[i] = valid ? radiusf : 0.0f;   // int radii emitted as float slots
}

// ------------- kernel 2: stable rank-sort by depth + scatter ---------
__global__ void gsplat_sort_scatter(const float* __restrict__ keys,
                                    const float* __restrict__ U,
                                    float* __restrict__ S)
{
    __shared__ float kl[GN];
    for (int j = threadIdx.x; j < GN; j += blockDim.x) kl[j] = keys[j];
    __syncthreads();

    int i = blockIdx.x * blockDim.x + threadIdx.x;
    if (i >= GN) return;
    float ki = kl[i];
    int rank = 0;
    for (int j = 0; j < GN; ++j) {
        float kj = kl[j];
        rank += (kj < ki) || ((kj == ki) && (j < i));
    }
    #pragma unroll
    for (int p = 0; p < 9; ++p)
        S[p*GN + rank] = U[p*GN + i];
}

// -------- kernel 3: per-pixel ordered blend, WMMA color einsum -------
// block = 128 threads = 4 wave32 = one image row. Each wave owns 32 px:
// each lane runs the ordered alpha/transmittance chain for its own pixel,
// then two V_WMMA_F32_16X16X4_F32 accumulate weights x colors for the
// wave's two 16-pixel row groups. A-operands are assembled with
// __shfl_xor(.,16) lane swaps; B (colors, zero-padded to 16 cols) is
// pre-laid-out in LDS in exact WMMA order -> one ds_load_b64, no branches.
__global__ void __launch_bounds__(128)
gsplat_blend(const float* __restrict__ S,
             const float* __restrict__ bg,
             float* __restrict__ out)
{
    constexpr int CH = 128;                 // gaussians staged per chunk
    __shared__ float lp[6 * CH];            // A,B,C,op,px,py
    __shared__ float bp[(CH / 4) * 64];     // B-matrix tiles, WMMA lane order
    __shared__ float tf[128];               // per-pixel final transmittance

    const int  lane = threadIdx.x & 31;
    const int  wave = threadIdx.x >> 5;     // 0..3
    const bool hi   = lane >= 16;
    const int  n    = lane & 15;            // D/B column (color channel)
    const int  y    = blockIdx.x;
    const int  px0  = wave * 32;            // wave's first pixel x
    const float xf  = (float)(px0 + lane);  // this lane's pixel
    const float yf  = (float)y;

    float T = 1.0f;
    v8f acc0 = {};                          // D rows = pixels px0+0..15
    v8f acc1 = {};                          // D rows = pixels px0+16..31

    for (int base = 0; base < GN; base += CH) {
        __syncthreads();
        for (int t = threadIdx.x; t < 6 * CH; t += blockDim.x) {
            int p = t / CH, e = t - p * CH;
            lp[t] = S[p * GN + base + e];
        }
        // colors in WMMA B layout: group q (4 gaussians), 64 floats:
        //   [h(=lane half)][col n][which] ; g = q*4 + h*2 + which
        for (int t = threadIdx.x; t < (CH / 4) * 64; t += blockDim.x) {
            int q     = t >> 6;
            int rem   = t & 63;
            int h     = rem >> 5;
            int col   = (rem >> 1) & 15;
            int which = rem & 1;
            int g     = q * 4 + h * 2 + which;
            float v   = 0.0f;
            if (col < 3) v = S[(6 + col) * GN + base + g];
            bp[t] = v;
        }
        if (base + CH < GN)                 // warm next chunk (global_prefetch_b8)
            __builtin_prefetch(&S[base + CH + threadIdx.x], 0, 1);
        __syncthreads();

        for (int k = 0; k < CH; k += 4) {
            // 4 independent alpha evaluations for this lane's pixel
            float al[4];
            #pragma unroll
            for (int u = 0; u < 4; ++u) {
                int g = k + u;
                float A  = lp[0*CH + g];
                float B  = lp[1*CH + g];
                float C  = lp[2*CH + g];
                float op = lp[3*CH + g];
                float dx = lp[4*CH + g] - xf;
                float dy = lp[5*CH + g] - yf;
                float power = -0.5f * (A*dx*dx + C*dy*dy) - B*dx*dy;
                float a = fminf(0.99f, op * __expf(power));
                al[u] = (power > 0.0f || a < (1.0f / 255.0f)) ? 0.0f : a;
            }
            // ordered transmittance chain (front-to-back)
            float w0 = al[0] * T;  T *= 1.0f - al[0];
            float w1 = al[1] * T;  T *= 1.0f - al[1];
            float w2 = al[2] * T;  T *= 1.0f - al[2];
            float w3 = al[3] * T;  T *= 1.0f - al[3];

            // swap halves so each WMMA sees the right pixels' weights
            float sx0 = __shfl_xor(w0, 16, 32);
            float sx1 = __shfl_xor(w1, 16, 32);
            float sx2 = __shfl_xor(w2, 16, 32);
            float sx3 = __shfl_xor(w3, 16, 32);

            // A (16x4): v0 = K0 | K2 per lane half, v1 = K1 | K3
            v2f A0, A1;
            A0.x = hi ? sx2 : w0;   A0.y = hi ? sx3 : w1;   // rows = px0..+15
            A1.x = hi ? w2 : sx0;   A1.y = hi ? w3 : sx1;   // rows = px0+16..+31

            // B (4x16): single ds_load_b64 in exact lane order
            const v2f Bm = *(const v2f*)&bp[(k << 4) + (hi ? 32 : 0) + (n << 1)];

            acc0 = __builtin_amdgcn_wmma_f32_16x16x4_f32(
                false, A0, false, Bm, (short)0, acc0, false, false);
            acc1 = __builtin_amdgcn_wmma_f32_16x16x4_f32(
                false, A1, false, Bm, (short)0, acc1, false, false);
        }
    }

    tf[wave * 32 + lane] = T;
    __syncthreads();

    if (n < 3) {                            // column n = color channel
        float bgc = bg[n];
        #pragma unroll
        for (int r = 0; r < 8; ++r) {
            int m  = hi ? (8 + r) : r;      // D vgpr r -> row m per lane half
            int xA = px0 + m;
            int xB = px0 + 16 + m;
            out[n * HW + y * IMG_W + xA] = acc0[r] + tf[wave * 32 + m]      * bgc;
            out[n * HW + y * IMG_W + xB] = acc1[r] + tf[wave * 32 + 16 + m] * bgc;
        }
    }
}

// ------------------------------ launch -------------------------------
extern "C" void kernel_launch(void* const* d_in, const int* in_sizes, int n_in,
                              void* d_out, int out_size, void* d_ws, size_t ws_size,
                              hipStream_t stream)
{
    const float* means3d = (const float*)d_in[0];
    // d_in[1] = means_2d (unused by reference math)
    const float* colors  = (const float*)d_in[2];
    const float* opac    = (const float*)d_in[3];
    const float* scales  = (const float*)d_in[4];
    const float* rots    = (const float*)d_in[5];
    const float* viewm   = (const float*)d_in[6];
    const float* projm   = (const float*)d_in[7];
    // d_in[8] = campos (unused)
    const float* bg      = (const float*)d_in[9];

    float* out  = (float*)d_out;            // [3*HW image | GN radii]
    float* ws   = (float*)d_ws;
    float* keys = ws;
    float* U    = ws + GN;
    float* S    = ws + 10 * GN;

    gsplat_preprocess<<<(GN + 255) / 256, 256, 0, stream>>>(
        means3d, colors, opac, scales, rots, viewm, projm, keys, U, out + 3 * HW);

    gsplat_sort_scatter<<<(GN + 255) / 256, 256, 0, stream>>>(keys, U, S);

    gsplat_blend<<<IMG_H, 128, 0, stream>>>(S, bg, out);
}